// HierarchicalNeuralTexture_86388972192291
// MI455X (gfx1250) — compile-verified
//
#include <hip/hip_runtime.h>

#define CLAMP_LO (-123.68f)
#define CLAMP_HI (151.061f)
#define TILE_X 256

static __device__ __forceinline__ float clampTex(float v) {
  return fminf(fmaxf(v, CLAMP_LO), CLAMP_HI);
}

// ---------------------------------------------------------------------------
// Pass 1: clamp + transpose  [16, N] -> [N, 16]  (N = H*W, multiple of 256).
// Stages a 16x256-float tile through LDS using the gfx1250 async global->LDS
// DMA path (ASYNCcnt), then writes clamped 64B-per-texel records.
// ---------------------------------------------------------------------------
__global__ void __launch_bounds__(256)
clamp_transpose_kernel(const float* __restrict__ src,
                       float* __restrict__ dst,
                       unsigned N) {
  __shared__ __align__(16) float tile[16][TILE_X];
  const unsigned tid = threadIdx.x;
  const unsigned ntiles = N / TILE_X;
  const unsigned ldsBase = (unsigned)(size_t)(&tile[0][0]);

  for (unsigned t = blockIdx.x; t < ntiles; t += gridDim.x) {
    const unsigned p0 = t * TILE_X;

    // 1024 async 16B chunks cover the 16x256 tile; 4 per thread.
#pragma unroll
    for (int k = 0; k < 4; ++k) {
      const unsigned q  = tid + (unsigned)k * 256u;   // chunk id 0..1023
      const unsigned c  = q >> 6;                     // channel 0..15
      const unsigned x4 = (q & 63u) << 2;             // texel-in-tile (mult of 4)
      const unsigned ldsOff = ldsBase + (c * (unsigned)TILE_X + x4) * 4u;
      const unsigned long long ga =
          (unsigned long long)(size_t)(src + (size_t)c * N + p0 + x4);
      asm volatile("global_load_async_to_lds_b128 %0, %1, off"
                   :: "v"(ldsOff), "v"(ga)
                   : "memory");
    }
    asm volatile("s_wait_asynccnt 0" ::: "memory");
    __syncthreads();

    // One texel per thread: gather 16 channels from LDS (conflict-free:
    // lane-varying column -> distinct banks), clamp, write 64B record.
    float4* __restrict__ o = (float4*)(dst + ((size_t)p0 + tid) * 16u);
#pragma unroll
    for (int g = 0; g < 4; ++g) {
      float4 v;
      v.x = clampTex(tile[4 * g + 0][tid]);
      v.y = clampTex(tile[4 * g + 1][tid]);
      v.z = clampTex(tile[4 * g + 2][tid]);
      v.w = clampTex(tile[4 * g + 3][tid]);
      o[g] = v;
    }
    __syncthreads();
  }
}

// ---------------------------------------------------------------------------
// Pass 2: bilinear sample (align_corners=True, border padding) from the
// transposed [N,16] textures; sum 4 levels; NT-store [B,C,Ho,Wo].
// ---------------------------------------------------------------------------
__global__ void __launch_bounds__(256)
sample_sum_kernel(const float2* __restrict__ grid,
                  const float* __restrict__ t0, const float* __restrict__ t1,
                  const float* __restrict__ t2, const float* __restrict__ t3,
                  float* __restrict__ out, int total, int HoWo) {
  const int idx = blockIdx.x * blockDim.x + threadIdx.x;
  if (idx >= total) return;
  const float2 uv = grid[idx];

  float4 acc[4];
#pragma unroll
  for (int g = 0; g < 4; ++g) acc[g] = make_float4(0.f, 0.f, 0.f, 0.f);

  const float* bases[4] = {t0, t1, t2, t3};
#pragma unroll
  for (int L = 0; L < 4; ++L) {
    const int W = 2048 >> L;           // square levels: H == W
    const float wm1 = (float)(W - 1);
    float ix = (uv.x + 1.0f) * 0.5f * wm1;
    float iy = (uv.y + 1.0f) * 0.5f * wm1;
    ix = fminf(fmaxf(ix, 0.0f), wm1);  // border padding: clamp continuous coords
    iy = fminf(fmaxf(iy, 0.0f), wm1);
    const float fx = floorf(ix), fy = floorf(iy);
    const float wx = ix - fx, wy = iy - fy;
    const int x0 = (int)fx, y0 = (int)fy;
    const int x1 = min(x0 + 1, W - 1);
    const int y1 = min(y0 + 1, W - 1);
    const float w00 = (1.f - wx) * (1.f - wy);
    const float w01 = wx * (1.f - wy);
    const float w10 = (1.f - wx) * wy;
    const float w11 = wx * wy;

    const float* bp = bases[L];
    const float4* p00 = (const float4*)(bp + ((size_t)y0 * W + x0) * 16u);
    const float4* p01 = (const float4*)(bp + ((size_t)y0 * W + x1) * 16u);
    const float4* p10 = (const float4*)(bp + ((size_t)y1 * W + x0) * 16u);
    const float4* p11 = (const float4*)(bp + ((size_t)y1 * W + x1) * 16u);
#pragma unroll
    for (int g = 0; g < 4; ++g) {
      const float4 a = p00[g], b = p01[g], c = p10[g], d = p11[g];
      acc[g].x = fmaf(w00, a.x, fmaf(w01, b.x, fmaf(w10, c.x, fmaf(w11, d.x, acc[g].x))));
      acc[g].y = fmaf(w00, a.y, fmaf(w01, b.y, fmaf(w10, c.y, fmaf(w11, d.y, acc[g].y))));
      acc[g].z = fmaf(w00, a.z, fmaf(w01, b.z, fmaf(w10, c.z, fmaf(w11, d.z, acc[g].z))));
      acc[g].w = fmaf(w00, a.w, fmaf(w01, b.w, fmaf(w10, c.w, fmaf(w11, d.w, acc[g].w))));
    }
  }

  const int b   = idx / HoWo;
  const int pix = idx - b * HoWo;
  const float accs[16] = {acc[0].x, acc[0].y, acc[0].z, acc[0].w,
                          acc[1].x, acc[1].y, acc[1].z, acc[1].w,
                          acc[2].x, acc[2].y, acc[2].z, acc[2].w,
                          acc[3].x, acc[3].y, acc[3].z, acc[3].w};
  float* op = out + (size_t)b * 16u * (size_t)HoWo + pix;
#pragma unroll
  for (int c = 0; c < 16; ++c)
    __builtin_nontemporal_store(accs[c], op + (size_t)c * HoWo);
}

// ---------------------------------------------------------------------------
// Fallback: sample straight from the [C,H,W] layout (clamp on the fly) if the
// workspace cannot hold the transposed textures.
// ---------------------------------------------------------------------------
__global__ void __launch_bounds__(256)
sample_direct_kernel(const float2* __restrict__ grid,
                     const float* __restrict__ t0, const float* __restrict__ t1,
                     const float* __restrict__ t2, const float* __restrict__ t3,
                     float* __restrict__ out, int total, int HoWo) {
  const int idx = blockIdx.x * blockDim.x + threadIdx.x;
  if (idx >= total) return;
  const float2 uv = grid[idx];

  float acc[16];
#pragma unroll
  for (int c = 0; c < 16; ++c) acc[c] = 0.f;

  const float* bases[4] = {t0, t1, t2, t3};
#pragma unroll
  for (int L = 0; L < 4; ++L) {
    const int W = 2048 >> L;
    const size_t N = (size_t)W * W;
    const float wm1 = (float)(W - 1);
    float ix = (uv.x + 1.0f) * 0.5f * wm1;
    float iy = (uv.y + 1.0f) * 0.5f * wm1;
    ix = fminf(fmaxf(ix, 0.0f), wm1);
    iy = fminf(fmaxf(iy, 0.0f), wm1);
    const float fx = floorf(ix), fy = floorf(iy);
    const float wx = ix - fx, wy = iy - fy;
    const int x0 = (int)fx, y0 = (int)fy;
    const int x1 = min(x0 + 1, W - 1);
    const int y1 = min(y0 + 1, W - 1);
    const float w00 = (1.f - wx) * (1.f - wy);
    const float w01 = wx * (1.f - wy);
    const float w10 = (1.f - wx) * wy;
    const float w11 = wx * wy;
    const size_t i00 = (size_t)y0 * W + x0, i01 = (size_t)y0 * W + x1;
    const size_t i10 = (size_t)y1 * W + x0, i11 = (size_t)y1 * W + x1;
    const float* bp = bases[L];
#pragma unroll
    for (int c = 0; c < 16; ++c) {
      const float* cp = bp + (size_t)c * N;
      const float a = clampTex(cp[i00]);
      const float bb = clampTex(cp[i01]);
      const float cc = clampTex(cp[i10]);
      const float dd = clampTex(cp[i11]);
      acc[c] = fmaf(w00, a, fmaf(w01, bb, fmaf(w10, cc, fmaf(w11, dd, acc[c]))));
    }
  }

  const int b   = idx / HoWo;
  const int pix = idx - b * HoWo;
  float* op = out + (size_t)b * 16u * (size_t)HoWo + pix;
#pragma unroll
  for (int c = 0; c < 16; ++c)
    __builtin_nontemporal_store(acc[c], op + (size_t)c * HoWo);
}

extern "C" void kernel_launch(void* const* d_in, const int* in_sizes, int n_in,
                              void* d_out, int out_size, void* d_ws, size_t ws_size,
                              hipStream_t stream) {
  const float2* x   = (const float2*)d_in[0];
  const float* tex0 = (const float*)d_in[1];
  const float* tex1 = (const float*)d_in[2];
  const float* tex2 = (const float*)d_in[3];
  const float* tex3 = (const float*)d_in[4];
  float* out = (float*)d_out;

  const int total = in_sizes[0] / 2;     // B*Ho*Wo = 4*960*960
  const int HoWo  = 960 * 960;

  const size_t n0 = 2048u * 2048u, n1 = 1024u * 1024u;
  const size_t n2 = 512u * 512u,   n3 = 256u * 256u;
  const size_t needBytes = (n0 + n1 + n2 + n3) * 16u * sizeof(float);

  const int blk = 256;
  const int sgrid = (total + blk - 1) / blk;

  if (ws_size >= needBytes) {
    float* w0 = (float*)d_ws;
    float* w1 = w0 + n0 * 16u;
    float* w2 = w1 + n1 * 16u;
    float* w3 = w2 + n2 * 16u;
    clamp_transpose_kernel<<<(unsigned)(n0 / TILE_X), blk, 0, stream>>>(tex0, w0, (unsigned)n0);
    clamp_transpose_kernel<<<(unsigned)(n1 / TILE_X), blk, 0, stream>>>(tex1, w1, (unsigned)n1);
    clamp_transpose_kernel<<<(unsigned)(n2 / TILE_X), blk, 0, stream>>>(tex2, w2, (unsigned)n2);
    clamp_transpose_kernel<<<(unsigned)(n3 / TILE_X), blk, 0, stream>>>(tex3, w3, (unsigned)n3);
    sample_sum_kernel<<<sgrid, blk, 0, stream>>>(x, w0, w1, w2, w3, out, total, HoWo);
  } else {
    sample_direct_kernel<<<sgrid, blk, 0, stream>>>(x, tex0, tex1, tex2, tex3, out, total, HoWo);
  }
}